// LanguageModelSRNN_48593259987425
// MI455X (gfx1250) — compile-verified
//
#include <hip/hip_runtime.h>
#include <math.h>

#define VOCAB  32000
#define HIDDEN 1024
#define BATCH  8
#define SEQ    256

typedef __attribute__((ext_vector_type(2))) float v2f;
typedef __attribute__((ext_vector_type(8))) float v8f;

// ------------------------------------------------------------------
// Kernel 1: sequential recurrence, parallel over (hidden_row, batch).
// 32 blocks x 256 threads = 8192 threads = 1024 rows * 8 batch.
// Grid-wide step barrier via device-scope atomic counter (32 WGs are
// always co-resident on MI455X).  Writes H_all[(b*SEQ+t)*HIDDEN + i].
// ------------------------------------------------------------------
__global__ __launch_bounds__(256) void srnn_recur(const int* __restrict__ x,
                                                  const float* __restrict__ U,
                                                  const float* __restrict__ W,
                                                  float* __restrict__ Hall,
                                                  unsigned* __restrict__ cnt)
{
  const int gid = blockIdx.x * 256 + threadIdx.x;   // 0..8191
  const int i = gid >> 3;                           // hidden row
  const int b = gid & 7;                            // batch index
  const float4* Urow = (const float4*)(U + (size_t)i * HIDDEN);

  for (int t = 0; t < SEQ; ++t) {
    const int tok = x[b * SEQ + t];
    float acc = W[(size_t)i * VOCAB + tok];         // gathered W column elem
    if (t > 0) {
      const float4* hp =
          (const float4*)(Hall + (size_t)(b * SEQ + (t - 1)) * HIDDEN);
      float s = 0.f;
      #pragma unroll 4
      for (int j = 0; j < HIDDEN / 4; ++j) {
        float4 u = Urow[j];
        float4 h = hp[j];
        s += u.x * h.x + u.y * h.y + u.z * h.z + u.w * h.w;
      }
      acc += s;
    }
    Hall[(size_t)(b * SEQ + t) * HIDDEN + i] = tanhf(acc);

    // ---- grid-wide step barrier ----
    __threadfence();
    __syncthreads();
    if (threadIdx.x == 0) {
      __hip_atomic_fetch_add(cnt, 1u, __ATOMIC_RELEASE, __HIP_MEMORY_SCOPE_AGENT);
      const unsigned target = (unsigned)gridDim.x * (unsigned)(t + 1);
      while (__hip_atomic_load(cnt, __ATOMIC_ACQUIRE, __HIP_MEMORY_SCOPE_AGENT) < target)
        __builtin_amdgcn_s_sleep(2);
    }
    __syncthreads();
  }
}

// ------------------------------------------------------------------
// Kernel 2: logits GEMM  Out[n, v] = sum_k Hall[n,k] * V[v,k]
// M = 2048, N = 32000, K = 1024.  WMMA f32 16x16x4.
// Block: 256 thr (8 waves) -> 16 rows x 256 vocab cols.
// A slab (16 x 1024 f32 = 64 KB) staged in LDS, k-major for
// conflict-free a-fragment reads.
// ------------------------------------------------------------------
__global__ __launch_bounds__(256) void srnn_logits(const float* __restrict__ Hall,
                                                   const float* __restrict__ Vm,
                                                   float* __restrict__ out)
{
  __shared__ float As[HIDDEN * 16];                 // As[k*16 + m], 64 KB
  const int row0 = blockIdx.y * 16;                 // M tile base (B*S rows)
  const int wave = threadIdx.x >> 5;
  const int lane = threadIdx.x & 31;
  const int v0   = blockIdx.x * 256 + wave * 32;    // this wave's vocab base
  const int lm   = lane & 15;
  const int kh   = lane >> 4;                       // 0 or 1

  // cooperative fill of A slab: global float4 reads, k-major LDS writes
  for (int it = 0; it < 16; ++it) {
    int idx = it * 256 + threadIdx.x;               // 0..4095 float4 units
    int m   = idx & 15;
    int k4  = idx >> 4;                             // 0..255
    float4 g = *(const float4*)(Hall + (size_t)(row0 + m) * HIDDEN + k4 * 4);
    As[(k4 * 4 + 0) * 16 + m] = g.x;
    As[(k4 * 4 + 1) * 16 + m] = g.y;
    As[(k4 * 4 + 2) * 16 + m] = g.z;
    As[(k4 * 4 + 3) * 16 + m] = g.w;
  }
  __syncthreads();

  // B-matrix lane pointers: B[k][n] = V[(v0+n)*HIDDEN + k], k = kb + 2*kh + {0,1}
  const float* Vp0 = Vm + (size_t)(v0 +      lm) * HIDDEN + 2 * kh;
  const float* Vp1 = Vm + (size_t)(v0 + 16 + lm) * HIDDEN + 2 * kh;

  v8f acc0 = {};
  v8f acc1 = {};
  for (int kb = 0; kb < HIDDEN; kb += 4) {
    v2f a;                                          // A[m=lm][kb + 2*kh + {0,1}]
    a.x = As[(kb + 2 * kh)     * 16 + lm];
    a.y = As[(kb + 2 * kh + 1) * 16 + lm];
    v2f b0 = *(const v2f*)(Vp0 + kb);
    v2f b1 = *(const v2f*)(Vp1 + kb);
    acc0 = __builtin_amdgcn_wmma_f32_16x16x4_f32(false, a, false, b0,
                                                 (short)0, acc0, false, false);
    acc1 = __builtin_amdgcn_wmma_f32_16x16x4_f32(false, a, false, b1,
                                                 (short)0, acc1, false, false);
  }

  // D layout: lane -> n = lm, vgpr r -> m = r + 8*kh
  #pragma unroll
  for (int r = 0; r < 8; ++r) {
    const size_t row = (size_t)(row0 + r + 8 * kh) * VOCAB;
    out[row + v0 +      lm] = acc0[r];
    out[row + v0 + 16 + lm] = acc1[r];
  }
}

// ------------------------------------------------------------------
// Kernel 3: row softmax in place.  One block per row (2048 rows of
// 32000).  Online max/sum, LDS tree reduce, then normalize pass
// (row stays L2-hot between the two passes).
// ------------------------------------------------------------------
__global__ __launch_bounds__(256) void srnn_softmax(float* __restrict__ out)
{
  __shared__ float smax[256];
  __shared__ float ssum[256];
  float* row = out + (size_t)blockIdx.x * VOCAB;

  float m = -INFINITY, s = 0.f;
  for (int j = threadIdx.x; j < VOCAB; j += 256) {
    float v  = row[j];
    float nm = fmaxf(m, v);
    s = s * __expf(m - nm) + __expf(v - nm);
    m = nm;
  }
  smax[threadIdx.x] = m;
  ssum[threadIdx.x] = s;
  __syncthreads();
  for (int off = 128; off > 0; off >>= 1) {
    if (threadIdx.x < off) {
      float m1 = smax[threadIdx.x], s1 = ssum[threadIdx.x];
      float m2 = smax[threadIdx.x + off], s2 = ssum[threadIdx.x + off];
      float nm = fmaxf(m1, m2);
      ssum[threadIdx.x] = s1 * __expf(m1 - nm) + s2 * __expf(m2 - nm);
      smax[threadIdx.x] = nm;
    }
    __syncthreads();
  }
  const float M    = smax[0];
  const float invS = 1.0f / ssum[0];
  for (int j = threadIdx.x; j < VOCAB; j += 256) {
    row[j] = __expf(row[j] - M) * invS;
  }
}

// ------------------------------------------------------------------
extern "C" void kernel_launch(void* const* d_in, const int* in_sizes, int n_in,
                              void* d_out, int out_size, void* d_ws, size_t ws_size,
                              hipStream_t stream) {
  (void)in_sizes; (void)n_in; (void)out_size; (void)ws_size;
  const int*   x  = (const int*)d_in[0];    // (8, 256) int32 tokens
  const float* U  = (const float*)d_in[1];  // (1024, 1024)
  const float* W  = (const float*)d_in[2];  // (1024, 32000)
  const float* Vm = (const float*)d_in[3];  // (32000, 1024)
  float* out = (float*)d_out;               // (8, 256, 32000)

  unsigned* cnt = (unsigned*)d_ws;                       // barrier counter
  float* Hall   = (float*)((char*)d_ws + 256);           // 2048 x 1024 f32 (8 MB)

  // zero the step-barrier counter (capturable async memset)
  hipMemsetAsync(d_ws, 0, 256, stream);

  srnn_recur<<<32, 256, 0, stream>>>(x, U, W, Hall, cnt);
  srnn_logits<<<dim3(125, 128), 256, 0, stream>>>(Hall, Vm, out);
  srnn_softmax<<<2048, 256, 0, stream>>>(out);
}